// AttentionLayer_27642409517435
// MI455X (gfx1250) — compile-verified
//
#include <hip/hip_runtime.h>
#include <math.h>

// Problem constants
#define BB 32
#define LL 4096
#define EE 512
#define HH 512
#define AA 256

typedef float v2f __attribute__((ext_vector_type(2)));
typedef float v8f __attribute__((ext_vector_type(8)));
typedef unsigned int u32x4 __attribute__((ext_vector_type(4)));
typedef int i32x4 __attribute__((ext_vector_type(4)));
typedef int i32x8 __attribute__((ext_vector_type(8)));

// LDS row stride for the 16x512 X strip: 512 + 4 pad dwords per row
// (TDM pad: 2 dwords after every 256 stored dwords -> row r, col k lives at
//  r*516 + k + 2*(k>=256)). 516 % 64 banks = 4 -> the 16 A-frag lanes hit
// 16 distinct banks.
#define XSTRIDE 516

// ---------------------------------------------------------------------------
// Kernel 0: transpose W_enc [E,A] -> W_encT [A,E] so WMMA B-fragments become
// contiguous b64 loads per lane. 32x32 LDS tiles, both sides coalesced.
// grid (E/32, A/32) x 256 threads.
// ---------------------------------------------------------------------------
__global__ __launch_bounds__(256) void transpose_kernel(
    const float* __restrict__ W_enc, float* __restrict__ W_encT) {
  __shared__ float t[32][33];
  const int kb = blockIdx.x * 32;  // E tile
  const int ab = blockIdx.y * 32;  // A tile
  const int lx = threadIdx.x & 31;
  const int ly = threadIdx.x >> 5;  // 8 rows per pass
  #pragma unroll
  for (int i = 0; i < 32; i += 8)
    t[ly + i][lx] = W_enc[(size_t)(kb + ly + i) * AA + ab + lx];
  __syncthreads();
  #pragma unroll
  for (int i = 0; i < 32; i += 8)
    W_encT[(size_t)(ab + ly + i) * EE + kb + lx] = t[lx][ly + i];
}

// ---------------------------------------------------------------------------
// Kernel 1: dec_attn[b,a] = decoder_hidden[b,:] @ W_dec[:,a] + b_dec[a]
// ---------------------------------------------------------------------------
__global__ __launch_bounds__(256) void dec_proj_kernel(
    const float* __restrict__ dh, const float* __restrict__ W_dec,
    const float* __restrict__ b_dec, float* __restrict__ dec_out) {
  const int b = blockIdx.x;
  const int a = threadIdx.x;  // 0..255
  float acc = b_dec[a];
  const float* dr = dh + b * HH;
  #pragma unroll 4
  for (int k = 0; k < HH; ++k) {
    acc = fmaf(dr[k], W_dec[k * AA + a], acc);
  }
  dec_out[b * AA + a] = acc;
}

// ---------------------------------------------------------------------------
// Kernel 2: fused enc projection + relu + W_v dot -> scores[B*L]
// One block per 16-row strip of (B*L). 256 threads = 8 waves.
// X strip staged to LDS by the Tensor Data Mover (wave 0), WMMA f32 16x16x4.
// ---------------------------------------------------------------------------
__global__ __launch_bounds__(256) void scores_kernel(
    const float* __restrict__ features, const float* __restrict__ W_encT,
    const float* __restrict__ b_enc, const float* __restrict__ dec_attn,
    const float* __restrict__ W_v, const float* __restrict__ b_v,
    float* __restrict__ scores) {
  __shared__ float xs[16 * XSTRIDE];
  __shared__ float red[8][16];

  const int tile = blockIdx.x;  // 0..8191
  const int r0 = tile << 4;     // first row of strip in B*L
  const int b = r0 >> 12;       // batch (L = 4096 rows per batch)
  const int tid = threadIdx.x;
  const int w = tid >> 5;  // wave id 0..7
  const int lane = tid & 31;

#if __has_builtin(__builtin_amdgcn_tensor_load_to_lds) && \
    __has_builtin(__builtin_amdgcn_s_wait_tensorcnt)
  if (w == 0) {
    // Tensor DMA descriptor: 2D tile 512x16 f32, row stride 512 elements,
    // LDS padding of 2 dwords every 256 dwords.
    const unsigned lds_base = (unsigned)(size_t)(&xs[0]);
    const unsigned long long ga =
        (unsigned long long)(size_t)(features + (size_t)r0 * EE);
    u32x4 g0;
    g0.x = 1u;  // count=1, user descriptor
    g0.y = lds_base;
    g0.z = (unsigned)ga;
    g0.w = (unsigned)((ga >> 32) & 0x01FFFFFFu) | 0x80000000u;  // type=2
    i32x8 g1;
    g1[0] = (2 << 16)     // data_size = 4B
          | (1 << 20)     // pad_enable
          | (7 << 22)     // pad_interval: every 256 dwords
          | (1 << 25);    // pad_amount: 2 dwords
    g1[1] = (EE & 0xFFFF) << 16;         // tensor_dim0[15:0] = 512
    g1[2] = (16 << 16);                  // tensor_dim0 hi=0 | tensor_dim1 lo=16
    g1[3] = (EE & 0xFFFF) << 16;         // tensor_dim1 hi=0 | tile_dim0 = 512
    g1[4] = 16;                          // tile_dim1 = 16, tile_dim2 = 0
    g1[5] = EE;                          // tensor_dim0_stride lo = 512
    g1[6] = 0;                           // stride hi | tensor_dim1_stride lo
    g1[7] = 0;                           // tensor_dim1_stride hi
    i32x4 gz4 = {};
    i32x8 gz8 = {};
    __builtin_amdgcn_tensor_load_to_lds(g0, g1, gz4, gz4, gz8, 0);
    __builtin_amdgcn_s_wait_tensorcnt(0);
  }
#else
  // Fallback: cooperative staging into the same padded layout.
  for (int i = tid; i < 16 * 128; i += 256) {
    const int row = i >> 7;
    const int c4 = (i & 127) << 2;
    const float4 v = *(const float4*)(features + (size_t)(r0 + row) * EE + c4);
    const int dst = row * XSTRIDE + c4 + ((c4 >= 256) ? 2 : 0);
    v2f lo, hi;
    lo.x = v.x; lo.y = v.y; hi.x = v.z; hi.y = v.w;
    *(v2f*)(&xs[dst]) = lo;
    *(v2f*)(&xs[dst + 2]) = hi;
  }
#endif
  __syncthreads();

  // Fragment indexing (wave32 WMMA f32 16x16x4):
  //   A 16x4:  M = lane&15, K = (lane>>4)*2 + vgpr
  //   B 4x16:  N = lane&15, K = (lane>>4)*2 + vgpr
  //   C 16x16: N = lane&15, M = (lane>>4)*8 + vgpr
  const int m = lane & 15;
  const int kh = (lane >> 4) << 1;
  const int n0c = (w << 5) + m;  // tile0 column for this lane
  const float* pA = &xs[m * XSTRIDE + kh];
  const float* pB0 = W_encT + (size_t)n0c * EE + kh;
  const float* pB1 = W_encT + (size_t)(n0c + 16) * EE + kh;

  v8f acc0 = {};
  v8f acc1 = {};

  // K half 0: k in [0,256)
  #pragma unroll 4
  for (int kb = 0; kb < 256; kb += 4) {
    const v2f a = *(const v2f*)(pA + kb);
    const v2f bf0 = *(const v2f*)(pB0 + kb);
    const v2f bf1 = *(const v2f*)(pB1 + kb);
    acc0 = __builtin_amdgcn_wmma_f32_16x16x4_f32(false, a, false, bf0,
                                                 (short)0, acc0, false, false);
    acc1 = __builtin_amdgcn_wmma_f32_16x16x4_f32(false, a, false, bf1,
                                                 (short)0, acc1, false, false);
  }
  // K half 1: k in [256,512); LDS has +2 dword pad after column 256
  const float* pA2 = pA + 258;
  const float* pB0b = pB0 + 256;
  const float* pB1b = pB1 + 256;
  #pragma unroll 4
  for (int kb = 0; kb < 256; kb += 4) {
    const v2f a = *(const v2f*)(pA2 + kb);
    const v2f bf0 = *(const v2f*)(pB0b + kb);
    const v2f bf1 = *(const v2f*)(pB1b + kb);
    acc0 = __builtin_amdgcn_wmma_f32_16x16x4_f32(false, a, false, bf0,
                                                 (short)0, acc0, false, false);
    acc1 = __builtin_amdgcn_wmma_f32_16x16x4_f32(false, a, false, bf1,
                                                 (short)0, acc1, false, false);
  }

  // Epilogue: relu(enc + b_enc + dec_attn) * W_v, reduce over A
  const float bias0 = b_enc[n0c] + dec_attn[b * AA + n0c];
  const float bias1 = b_enc[n0c + 16] + dec_attn[b * AA + n0c + 16];
  const float wv0 = W_v[n0c];
  const float wv1 = W_v[n0c + 16];

  float part[8];
  #pragma unroll
  for (int v = 0; v < 8; ++v) {
    float t0 = fmaxf(acc0[v] + bias0, 0.0f) * wv0;
    float t1 = fmaxf(acc1[v] + bias1, 0.0f) * wv1;
    part[v] = t0 + t1;
  }
  // Reduce across the 16 lanes holding N=0..15 (xor within 16-lane halves)
  #pragma unroll
  for (int mask = 1; mask <= 8; mask <<= 1) {
    #pragma unroll
    for (int v = 0; v < 8; ++v) part[v] += __shfl_xor(part[v], mask, 32);
  }
  if (m == 0) {  // lanes 0 (rows 0..7) and 16 (rows 8..15)
    const int rb = (lane >> 4) << 3;
    #pragma unroll
    for (int v = 0; v < 8; ++v) red[w][rb + v] = part[v];
  }
  __syncthreads();

  if (tid < 16) {
    float s = b_v[0];
    #pragma unroll
    for (int ww = 0; ww < 8; ++ww) s += red[ww][tid];
    scores[tile * 16 + tid] = s;
  }
}

// ---------------------------------------------------------------------------
// Kernel 3: softmax over L per batch. One block per batch, 256 threads,
// 16 elements per thread held in registers.
// ---------------------------------------------------------------------------
__global__ __launch_bounds__(256) void softmax_kernel(
    const float* __restrict__ scores, float* __restrict__ wout) {
  __shared__ float s1[256];
  const int b = blockIdx.x;
  const int tid = threadIdx.x;
  const float* sp = scores + b * LL;

  float v[16];
  float mx = -INFINITY;
  #pragma unroll
  for (int i = 0; i < 16; ++i) {
    v[i] = sp[i * 256 + tid];
    mx = fmaxf(mx, v[i]);
  }
  s1[tid] = mx;
  __syncthreads();
  for (int off = 128; off > 0; off >>= 1) {
    if (tid < off) s1[tid] = fmaxf(s1[tid], s1[tid + off]);
    __syncthreads();
  }
  const float M = s1[0];
  __syncthreads();

  float loc = 0.0f;
  #pragma unroll
  for (int i = 0; i < 16; ++i) {
    v[i] = expf(v[i] - M);
    loc += v[i];
  }
  s1[tid] = loc;
  __syncthreads();
  for (int off = 128; off > 0; off >>= 1) {
    if (tid < off) s1[tid] += s1[tid + off];
    __syncthreads();
  }
  const float inv = 1.0f / s1[0];

  float* wp = wout + b * LL;
  #pragma unroll
  for (int i = 0; i < 16; ++i) wp[i * 256 + tid] = v[i] * inv;
}

// ---------------------------------------------------------------------------
// Kernel 4a: partial weighted sums over 512-row L-chunks (no atomics).
// grid (B, E/256, 8); thread owns one e column within the chunk.
// ---------------------------------------------------------------------------
__global__ __launch_bounds__(256) void wsum_partial_kernel(
    const float* __restrict__ features, const float* __restrict__ wts,
    float* __restrict__ part) {
  __shared__ float wl[512];
  const int b = blockIdx.x;
  const int e = blockIdx.y * 256 + threadIdx.x;
  const int lc = blockIdx.z;

  const float* wp = wts + b * LL + lc * 512;
  wl[threadIdx.x] = wp[threadIdx.x];
  wl[threadIdx.x + 256] = wp[threadIdx.x + 256];
  __syncthreads();

  const float* f = features + ((size_t)b * LL + (size_t)lc * 512) * EE + e;
  float acc = 0.0f;
  #pragma unroll 8
  for (int l = 0; l < 512; ++l) {
    acc = fmaf(f[(size_t)l * EE], wl[l], acc);
  }
  part[lc * (BB * EE) + b * EE + e] = acc;
}

// ---------------------------------------------------------------------------
// Kernel 4b: reduce 8 partials -> output[B,E]
// ---------------------------------------------------------------------------
__global__ __launch_bounds__(256) void wsum_reduce_kernel(
    const float* __restrict__ part, float* __restrict__ out) {
  const int i = blockIdx.x * 256 + threadIdx.x;  // 0..16383
  float s = 0.0f;
  #pragma unroll
  for (int c = 0; c < 8; ++c) s += part[c * (BB * EE) + i];
  out[i] = s;
}

// ---------------------------------------------------------------------------
extern "C" void kernel_launch(void* const* d_in, const int* in_sizes, int n_in,
                              void* d_out, int out_size, void* d_ws, size_t ws_size,
                              hipStream_t stream) {
  const float* features = (const float*)d_in[0];  // [B,L,E]
  const float* dec_hid  = (const float*)d_in[1];  // [B,H]
  const float* W_enc    = (const float*)d_in[2];  // [E,A]
  const float* b_enc    = (const float*)d_in[3];  // [A]
  const float* W_dec    = (const float*)d_in[4];  // [H,A]
  const float* b_dec    = (const float*)d_in[5];  // [A]
  const float* W_v      = (const float*)d_in[6];  // [A,1]
  const float* b_v      = (const float*)d_in[7];  // [1]

  float* out      = (float*)d_out;            // [B*E] output
  float* attn_out = (float*)d_out + BB * EE;  // [B*L] attn weights

  float* ws      = (float*)d_ws;
  float* dec_ws  = ws;                    // B*A   =   8192 floats
  float* scores  = ws + BB * AA;          // B*L   = 131072 floats
  float* partial = scores + BB * LL;      // 8*B*E = 131072 floats
  float* W_encT  = partial + 8 * BB * EE; // A*E   = 131072 floats

  transpose_kernel<<<dim3(EE / 32, AA / 32), 256, 0, stream>>>(W_enc, W_encT);

  dec_proj_kernel<<<BB, 256, 0, stream>>>(dec_hid, W_dec, b_dec, dec_ws);

  scores_kernel<<<(BB * LL) / 16, 256, 0, stream>>>(
      features, W_encT, b_enc, dec_ws, W_v, b_v, scores);

  softmax_kernel<<<BB, 256, 0, stream>>>(scores, attn_out);

  wsum_partial_kernel<<<dim3(BB, EE / 256, 8), 256, 0, stream>>>(
      features, attn_out, partial);

  wsum_reduce_kernel<<<(BB * EE) / 256, 256, 0, stream>>>(partial, out);
}